// BPKD_24979529793863
// MI455X (gfx1250) — compile-verified
//
#include <hip/hip_runtime.h>
#include <hip/hip_bf16.h>

// ---------------------------------------------------------------------------
// BPKD loss (edge/body masked spatial-softmax KL) for MI455X / gfx1250.
// Bandwidth-bound: ~113MB of reads -> ~5us roofline @ 23.3 TB/s.
// Uses CDNA5 Tensor Data Mover (tensor_load_to_lds) to double-buffer the
// student/teacher logit streams into LDS, gated by TENSORcnt.
// ---------------------------------------------------------------------------

#define Bn 4
#define Cn 14
#define Hn 512
#define Wn 512
#define HWn (Hn * Wn)                 // 262144
#define NPAIR (Bn * (Cn - 1))         // 52
#define TILE 512                      // f32 elements per TDM tile (2KB)
#define WAVES 4                       // 128 threads / wave32
#define ITERS 4                       // tiles per wave
#define PIX_PER_WAVE (TILE * ITERS)   // 2048
#define PIX_PER_BLOCK (PIX_PER_WAVE * WAVES) // 8192
#define NCHUNK (HWn / PIX_PER_BLOCK)  // 32
#define NEGI (-1.0e30f)

typedef unsigned int u32;
typedef u32 v4u __attribute__((ext_vector_type(4)));
typedef int v8i __attribute__((ext_vector_type(8)));
typedef int v4i __attribute__((ext_vector_type(4)));

// Issue a TDM DMA of `nelem` f32 (one row) from global memory to LDS.
// D# packing per CDNA5 ISA 8.3/8.4: group0 = {flags, lds_addr, gaddr lo,
// gaddr hi | type=2}; group1 = data_size=4B, tensor_dim0 = tile_dim0 = nelem,
// tensor_dim1 = tile_dim1 = 1, stride0 = nelem. Trailing groups zero (<=2D).
__device__ __forceinline__ void tdm_load_row(u32 lds_addr, const float* gsrc,
                                             u32 nelem) {
  unsigned long long ga = (unsigned long long)(size_t)gsrc;
  u32 lds_s = __builtin_amdgcn_readfirstlane(lds_addr);
  u32 ga_lo = __builtin_amdgcn_readfirstlane((u32)ga);
  u32 ga_hi = __builtin_amdgcn_readfirstlane((u32)(ga >> 32));
  v4u g0;
  g0.x = 1u;                                   // count=1 (valid), user mode
  g0.y = lds_s;                                // lds_addr (bytes)
  g0.z = ga_lo;                                // global_addr[31:0]
  g0.w = (ga_hi & 0x1FFFFFFu) | (2u << 30);    // global_addr[56:32] | type=2
  v8i g1;
  g1[0] = (int)(2u << 16);                     // data_size = 2 -> 4 bytes
  g1[1] = (int)((nelem & 0xFFFFu) << 16);      // tensor_dim0[15:0]
  g1[2] = (int)(((nelem >> 16) & 0xFFFFu) | (1u << 16)); // td0 hi | td1=1 lo
  g1[3] = (int)((nelem & 0xFFFFu) << 16);      // td1 hi (0) | tile_dim0
  g1[4] = 1;                                   // tile_dim1=1, tile_dim2=0
  g1[5] = (int)nelem;                          // tensor_dim0_stride[31:0]
  g1[6] = 0;
  g1[7] = 0;
  v4i z4 = {0, 0, 0, 0};
  v8i z8 = {0, 0, 0, 0, 0, 0, 0, 0};
  __builtin_amdgcn_tensor_load_to_lds(g0, g1, z4, z4, z8, 0);
}

// Online-softmax scalar updates / combines (max-rescaled).
__device__ __forceinline__ void upd(float x, float& m, float& Z) {
  if (x > m) { Z = Z * __expf(m - x) + 1.0f; m = x; }
  else       { Z += __expf(x - m); }
}
__device__ __forceinline__ void updW(float y, float d, float& m, float& Z,
                                     float& W) {
  if (y > m) { float r = __expf(m - y); Z = Z * r + 1.0f; W = W * r + d; m = y; }
  else       { float e = __expf(y - m); Z += e; W += e * d; }
}
__device__ __forceinline__ void combine(float& m, float& Z, float om, float oZ) {
  float M = fmaxf(m, om);
  Z = Z * __expf(m - M) + oZ * __expf(om - M);
  m = M;
}
__device__ __forceinline__ void combineW(float& m, float& Z, float& W, float om,
                                         float oZ, float oW) {
  float M = fmaxf(m, om);
  float a = __expf(m - M), b = __expf(om - M);
  Z = Z * a + oZ * b;
  W = W * a + oW * b;
  m = M;
}

__global__ __launch_bounds__(128) void bpkd_stage1(
    const float* __restrict__ S, const float* __restrict__ T,
    const int* __restrict__ L, float* __restrict__ part) {
  __shared__ float tile[WAVES][2][2][TILE];  // [wave][dblbuf][S/T][elem] 32KB
  __shared__ float red[WAVES][11];

  const int bx    = blockIdx.x;
  const int pair  = bx >> 5;             // / NCHUNK
  const int chunk = bx & (NCHUNK - 1);
  const int b     = pair / (Cn - 1);
  const int cc    = pair - b * (Cn - 1);
  const int cls   = cc + 1;              // classes 1..13
  const int tid   = threadIdx.x;
  const int wave  = tid >> 5;
  const int lane  = tid & 31;

  const float* Sb = S + (size_t)(b * Cn + cls) * HWn;
  const float* Tb = T + (size_t)(b * Cn + cls) * HWn;
  const int*   Lb = L + (size_t)b * HWn;

  const int waveBase = chunk * PIX_PER_BLOCK + wave * PIX_PER_WAVE;

  float mxe = NEGI, zxe = 0.f, mye = NEGI, zye = 0.f, we = 0.f, ce = 0.f;
  float mxb = NEGI, zxb = 0.f, myb = NEGI, zyb = 0.f, wb = 0.f;

  // Prime the pipeline: iter 0 tiles for S and T.
  tdm_load_row((u32)(size_t)&tile[wave][0][0][0], Sb + waveBase, TILE);
  tdm_load_row((u32)(size_t)&tile[wave][0][1][0], Tb + waveBase, TILE);

#pragma unroll
  for (int it = 0; it < ITERS; ++it) {
    const int pb = it & 1;
    if (it + 1 < ITERS) {
      const int nb2 = (it + 1) & 1;
      tdm_load_row((u32)(size_t)&tile[wave][nb2][0][0],
                   Sb + waveBase + (it + 1) * TILE, TILE);
      tdm_load_row((u32)(size_t)&tile[wave][nb2][1][0],
                   Tb + waveBase + (it + 1) * TILE, TILE);
      __builtin_amdgcn_s_wait_tensorcnt(2);  // current iter's 2 DMAs done
    } else {
      __builtin_amdgcn_s_wait_tensorcnt(0);
    }
    asm volatile("" ::: "memory");

    const int tileBase = waveBase + it * TILE;
#pragma unroll 4
    for (int j = 0; j < TILE / 32; ++j) {
      const int e = j * 32 + lane;
      const int p = tileBase + e;
      float s = tile[wave][pb][0][e];
      float t = tile[wave][pb][1][e];
      const int hh = p >> 9;
      const int wp = p & (Wn - 1);
      int l0 = Lb[p];
      int lu = (hh > 0)      ? Lb[p - Wn] : -1;
      int ld = (hh < Hn - 1) ? Lb[p + Wn] : -1;
      int ll = (wp > 0)      ? Lb[p - 1]  : -1;
      int lr = (wp < Wn - 1) ? Lb[p + 1]  : -1;
      bool ero = (l0 == cls) & (lu == cls) & (ld == cls) & (ll == cls) &
                 (lr == cls);                                   // body
      bool dil = (l0 == cls) | (lu == cls) | (ld == cls) | (ll == cls) |
                 (lr == cls);
      bool edge = dil && !ero;
      float xe = edge ? s : 0.0f, ye = edge ? t : 0.0f;
      float xb = ero ? s : 0.0f,  yb = ero ? t : 0.0f;
      ce += edge ? 1.0f : 0.0f;
      upd(xe, mxe, zxe);
      updW(ye, ye - xe, mye, zye, we);
      upd(xb, mxb, zxb);
      updW(yb, yb - xb, myb, zyb, wb);
    }
    asm volatile("" ::: "memory");
  }

  // wave32 butterfly reduction
#pragma unroll
  for (int off = 16; off >= 1; off >>= 1) {
    float om, oz, ow;
    om = __shfl_xor(mxe, off, 32); oz = __shfl_xor(zxe, off, 32);
    combine(mxe, zxe, om, oz);
    om = __shfl_xor(mye, off, 32); oz = __shfl_xor(zye, off, 32);
    ow = __shfl_xor(we, off, 32);
    combineW(mye, zye, we, om, oz, ow);
    om = __shfl_xor(mxb, off, 32); oz = __shfl_xor(zxb, off, 32);
    combine(mxb, zxb, om, oz);
    om = __shfl_xor(myb, off, 32); oz = __shfl_xor(zyb, off, 32);
    ow = __shfl_xor(wb, off, 32);
    combineW(myb, zyb, wb, om, oz, ow);
    ce += __shfl_xor(ce, off, 32);
  }
  if (lane == 0) {
    red[wave][0] = mxe; red[wave][1] = zxe; red[wave][2] = mye;
    red[wave][3] = zye; red[wave][4] = we;  red[wave][5] = mxb;
    red[wave][6] = zxb; red[wave][7] = myb; red[wave][8] = zyb;
    red[wave][9] = wb;  red[wave][10] = ce;
  }
  __syncthreads();
  if (tid == 0) {
    float a0 = red[0][0], a1 = red[0][1], a2 = red[0][2], a3 = red[0][3];
    float a4 = red[0][4], a5 = red[0][5], a6 = red[0][6], a7 = red[0][7];
    float a8 = red[0][8], a9 = red[0][9], a10 = red[0][10];
    for (int wv = 1; wv < WAVES; ++wv) {
      combine(a0, a1, red[wv][0], red[wv][1]);
      combineW(a2, a3, a4, red[wv][2], red[wv][3], red[wv][4]);
      combine(a5, a6, red[wv][5], red[wv][6]);
      combineW(a7, a8, a9, red[wv][7], red[wv][8], red[wv][9]);
      a10 += red[wv][10];
    }
    float* o = part + (size_t)bx * 11;
    o[0] = a0; o[1] = a1; o[2] = a2; o[3] = a3; o[4] = a4; o[5] = a5;
    o[6] = a6; o[7] = a7; o[8] = a8; o[9] = a9; o[10] = a10;
  }
}

__global__ __launch_bounds__(64) void bpkd_stage2(const float* __restrict__ part,
                                                  float* __restrict__ out) {
  __shared__ float klE[NPAIR], klB[NPAIR], cntE[NPAIR];
  const int t = threadIdx.x;
  if (t < NPAIR) {
    float a0 = NEGI, a1 = 0.f, a2 = NEGI, a3 = 0.f, a4 = 0.f;
    float a5 = NEGI, a6 = 0.f, a7 = NEGI, a8 = 0.f, a9 = 0.f, a10 = 0.f;
    for (int ch = 0; ch < NCHUNK; ++ch) {
      const float* p = part + (size_t)(t * NCHUNK + ch) * 11;
      combine(a0, a1, p[0], p[1]);
      combineW(a2, a3, a4, p[2], p[3], p[4]);
      combine(a5, a6, p[5], p[6]);
      combineW(a7, a8, a9, p[7], p[8], p[9]);
      a10 += p[10];
    }
    // KL = W/Z_y - (lse_y - lse_x),  lse = m + log Z  (TEMP=1)
    float kle = a4 / a3 - ((a2 + __logf(a3)) - (a0 + __logf(a1)));
    float klb = a9 / a8 - ((a7 + __logf(a8)) - (a5 + __logf(a6)));
    klE[t] = kle; klB[t] = klb; cntE[t] = a10;
  }
  __syncthreads();
  if (t == 0) {
    float loss_e = 0.f, loss_b = 0.f;
    for (int b = 0; b < Bn; ++b) {
      float ne = 0.f, lei = 0.f;
      for (int cc = 0; cc < Cn - 1; ++cc) {
        const int i = b * (Cn - 1) + cc;
        if (cntE[i] > 0.f) {      // valid = edge_cnt > 0
          ne += cntE[i];
          lei += klE[i];
          loss_b += klB[i];
        }
      }
      if (lei > 0.f) loss_e += lei / fmaxf(ne, 1.0f);
    }
    out[0] = 50.0f * loss_e / (float)Bn;           // LOSS_W*EDGE_W/nb
    out[1] = 20.0f * loss_b / (float)(Cn * Bn);    // LOSS_W*BODY_W/(C*nb)
  }
}

extern "C" void kernel_launch(void* const* d_in, const int* in_sizes, int n_in,
                              void* d_out, int out_size, void* d_ws,
                              size_t ws_size, hipStream_t stream) {
  const float* S = (const float*)d_in[0];
  const float* T = (const float*)d_in[1];
  const int*   L = (const int*)d_in[2];
  float* out  = (float*)d_out;
  float* part = (float*)d_ws;  // NPAIR*NCHUNK*11 floats = 73,216 bytes

  bpkd_stage1<<<dim3(NPAIR * NCHUNK), dim3(128), 0, stream>>>(S, T, L, part);
  bpkd_stage2<<<dim3(1), dim3(64), 0, stream>>>(part, out);
}